// UpSampling_45019847197062
// MI455X (gfx1250) — compile-verified
//
#include <hip/hip_runtime.h>
#include <hip/hip_bf16.h>

// Memory-bound gather/average ("edge midpoint upsampling").
//   out[r, 0:N]        = 0
//   out[r, N+e]        = 0.5f * (data[r, edges[e,0]] + data[r, edges[e,1]])
// r ranges over B*C rows. Pure bandwidth problem (~820 MB HBM traffic,
// ~35us floor at 23.3 TB/s). Strategy:
//   - b128 non-temporal stores for the streamed output (keep L2 for the
//     gather source + edge list),
//   - int4 (32B-aligned) edge loads: 2 edges per load,
//   - global_prefetch on the edge stream one block-stride ahead.

typedef __attribute__((ext_vector_type(4))) float float4v;
typedef __attribute__((ext_vector_type(4))) int   int4v;

__global__ void upsample_mid_vec4(const float* __restrict__ data,
                                  const int*   __restrict__ edges,
                                  float*       __restrict__ out,
                                  int N, int E, int q4PerRow)
{
    const int q4  = blockIdx.x * blockDim.x + threadIdx.x;
    const int row = blockIdx.y;
    if (q4 >= q4PerRow) return;

    const int cols = N + E;
    float* __restrict__ orow = out + (size_t)row * (size_t)cols;
    const int j = q4 * 4;                       // first of 4 output columns

    if (j < N) {
        // Zero head: stream zeros, bypass-friendly NT store.
        float4v z = {0.0f, 0.0f, 0.0f, 0.0f};
        __builtin_nontemporal_store(z, (float4v*)(orow + j));
    } else {
        const int e = j - N;                    // multiple of 4
        // Prefetch the edge words one block-stride ahead (gfx1250
        // global_prefetch_b8). Address may run past the array end;
        // prefetches are speculative and side-effect free.
        __builtin_prefetch(edges + 2 * (e + (int)blockDim.x * 4), 0, 3);

        // 4 edges = 8 ints = two 16B loads, 32B-aligned (e % 4 == 0).
        const int4v* __restrict__ ep = (const int4v*)(edges + 2 * (size_t)e);
        const int4v e01 = ep[0];                // edges e, e+1
        const int4v e23 = ep[1];                // edges e+2, e+3

        const float* __restrict__ drow = data + (size_t)row * (size_t)N;

        // 8 scattered b32 gathers (L2-resident working set: 400KB/row).
        const float a0 = drow[e01.x], b0 = drow[e01.y];
        const float a1 = drow[e01.z], b1 = drow[e01.w];
        const float a2 = drow[e23.x], b2 = drow[e23.y];
        const float a3 = drow[e23.z], b3 = drow[e23.w];

        float4v m;
        m.x = (a0 + b0) * 0.5f;
        m.y = (a1 + b1) * 0.5f;
        m.z = (a2 + b2) * 0.5f;
        m.w = (a3 + b3) * 0.5f;
        __builtin_nontemporal_store(m, (float4v*)(orow + j));
    }
}

// Generic scalar fallback (only used if vector alignment doesn't hold).
__global__ void upsample_mid_scalar(const float* __restrict__ data,
                                    const int*   __restrict__ edges,
                                    float*       __restrict__ out,
                                    int N, int E)
{
    const int j   = blockIdx.x * blockDim.x + threadIdx.x;
    const int row = blockIdx.y;
    const int cols = N + E;
    if (j >= cols) return;

    float* __restrict__ orow = out + (size_t)row * (size_t)cols;
    if (j < N) {
        __builtin_nontemporal_store(0.0f, orow + j);
    } else {
        const int e  = j - N;
        const int p0 = edges[2 * (size_t)e + 0];
        const int p1 = edges[2 * (size_t)e + 1];
        const float* __restrict__ drow = data + (size_t)row * (size_t)N;
        __builtin_nontemporal_store((drow[p0] + drow[p1]) * 0.5f, orow + j);
    }
}

extern "C" void kernel_launch(void* const* d_in, const int* in_sizes, int n_in,
                              void* d_out, int out_size, void* d_ws, size_t ws_size,
                              hipStream_t stream) {
    const float* data  = (const float*)d_in[0];   // [rows, N] flat
    const int*   edges = (const int*)d_in[1];     // [E, 2]   flat
    float*       out   = (float*)d_out;           // [rows, N+E] flat

    // Derive geometry: rows*N = |data|, |edges| = 2E, rows*(N+E) = out_size.
    const int E    = in_sizes[1] / 2;
    const int rows = (int)(((long long)out_size - (long long)in_sizes[0]) / E);
    const int N    = in_sizes[0] / rows;
    const int cols = N + E;

    const int threads = 256;  // 8 wave32 waves per block

    if ((N % 4 == 0) && (cols % 4 == 0)) {
        const int q4PerRow = cols / 4;
        dim3 grid((q4PerRow + threads - 1) / threads, rows);
        upsample_mid_vec4<<<grid, dim3(threads), 0, stream>>>(
            data, edges, out, N, E, q4PerRow);
    } else {
        dim3 grid((cols + threads - 1) / threads, rows);
        upsample_mid_scalar<<<grid, dim3(threads), 0, stream>>>(
            data, edges, out, N, E);
    }
}